// LiquidHashRouter_37658273251870
// MI455X (gfx1250) — compile-verified
//
#include <hip/hip_runtime.h>
#include <hip/hip_bf16.h>
#include <math.h>

typedef __attribute__((ext_vector_type(2))) float v2f;
typedef __attribute__((ext_vector_type(4))) float v4f;
typedef __attribute__((ext_vector_type(8))) float v8f;

// ---------------------------------------------------------------------------
// Pass 1: routing. One wave (32 threads) per 16-token tile.
// Computes hash_codes tile = x_tile(16xD) @ W^T(Dx16) with V_WMMA_F32_16X16X4_F32,
// row-sums it, derives expert idx, evaluates the 8-term sine modulation,
// writes per-token scale = 1 + 0.1*mod into d_ws.
// ---------------------------------------------------------------------------
__global__ __launch_bounds__(32)
void lhr_route_kernel(const float* __restrict__ x,
                      const float* __restrict__ W,
                      const float* __restrict__ efreq,
                      const float* __restrict__ eamp,
                      float* __restrict__ scale,
                      int D, int T, int E)
{
    const int lane = threadIdx.x;        // 0..31
    const int half = lane >> 4;          // 0: K=0,1   1: K=2,3
    const int lrow = lane & 15;
    const long tok_base = (long)blockIdx.x * 16;

    // A fragment source: row = token, cols k.. ; per ISA A-layout (16x4 f32):
    //   VGPR0 = K=(half*2), VGPR1 = K=(half*2+1), M = lane%16
    const float* xrow = x + (tok_base + lrow) * (long)D + half * 2;
    // B fragment source: B[k][n] = W[n][k]; per B-layout N = lane%16,
    //   VGPR0 = K=(half*2), VGPR1 = K=(half*2+1)
    const float* wrow = W + (long)lrow * D + half * 2;

    v8f acc0 = {};
    v8f acc1 = {};
    #pragma unroll 4
    for (int k = 0; k < D; k += 8) {
        v2f a0 = *(const v2f*)(xrow + k);
        v2f b0 = *(const v2f*)(wrow + k);
        v2f a1 = *(const v2f*)(xrow + k + 4);
        v2f b1 = *(const v2f*)(wrow + k + 4);
        acc0 = __builtin_amdgcn_wmma_f32_16x16x4_f32(
                   false, a0, false, b0, (short)0, acc0, false, false);
        acc1 = __builtin_amdgcn_wmma_f32_16x16x4_f32(
                   false, a1, false, b1, (short)0, acc1, false, false);
    }

    // C/D layout: VGPR j, lanes 0-15 -> (M=j, N=lane), lanes 16-31 -> (M=j+8, N=lane-16).
    // Sum over N = 16-lane reduction within each half (xor masks 1,2,4,8 stay in-half).
    float r[8];
    #pragma unroll
    for (int j = 0; j < 8; ++j) {
        float v = acc0[j] + acc1[j];
        v += __shfl_xor(v, 1, 32);
        v += __shfl_xor(v, 2, 32);
        v += __shfl_xor(v, 4, 32);
        v += __shfl_xor(v, 8, 32);
        r[j] = v;
    }

    __shared__ float rs[16];
    if (lrow == 0) {
        #pragma unroll
        for (int j = 0; j < 8; ++j) rs[half * 8 + j] = r[j];
    }
    __syncthreads();

    if (lane < 16) {
        float hsum = rs[lane];                    // sum_h hash_codes[token][h]
        float s = fmodf(fabsf(hsum), (float)E);   // |.| % E, in [0, E)
        int idx = (int)s;
        idx &= (E - 1);                           // E=32 (pow2) safety clamp
        const long tok = tok_base + lane;
        const int tpos = (int)(tok % (long)T);    // position within sequence
        const float tn = (float)tpos * (6.283185307179586f / (float)T);
        const float* fr = efreq + idx * 8;
        const float* am = eamp  + idx * 8;
        float m = 0.0f;
        #pragma unroll
        for (int j = 0; j < 8; ++j)
            m += am[j] * __sinf(fr[j] * tn);
        scale[tok] = 1.0f + 0.1f * m;
    }
}

// ---------------------------------------------------------------------------
// Pass 2: out = x * scale[token], float4-vectorized streaming pass.
// x re-read hits L2 (134 MB < 192 MB L2); HBM traffic = read x + write out.
// ---------------------------------------------------------------------------
__global__ __launch_bounds__(256)
void lhr_apply_kernel(const float* __restrict__ x,
                      const float* __restrict__ scale,
                      float* __restrict__ out,
                      long n4, int shift /* log2(D/4) */)
{
    long i = (long)blockIdx.x * blockDim.x + threadIdx.x;
    if (i >= n4) return;
    v4f xv = ((const v4f*)x)[i];
    float s = scale[i >> shift];
    ((v4f*)out)[i] = xv * s;
}

extern "C" void kernel_launch(void* const* d_in, const int* in_sizes, int n_in,
                              void* d_out, int out_size, void* d_ws, size_t ws_size,
                              hipStream_t stream) {
    const float* x  = (const float*)d_in[0];   // (B,T,D) fp32
    const float* W  = (const float*)d_in[1];   // (16,D)  fp32
    const float* fr = (const float*)d_in[2];   // (E,8)   fp32
    const float* am = (const float*)d_in[3];   // (E,8)   fp32
    float* out = (float*)d_out;
    float* scale = (float*)d_ws;               // BT floats of scratch

    const int H = 16;
    const int D = in_sizes[1] / H;             // 2048
    const long total = (long)in_sizes[0];      // B*T*D
    const long BT = total / D;                 // 16384 tokens
    const int T = 4096;                        // sequence length (reference constant)
    const int E = in_sizes[2] / 8;             // 32 experts

    // Pass 1: one wave per 16-token tile.
    const int tiles = (int)(BT / 16);
    lhr_route_kernel<<<tiles, 32, 0, stream>>>(x, W, fr, am, scale, D, T, E);

    // Pass 2: vectorized scale.
    const long n4 = total / 4;
    int shift = 0;
    { int dq = D / 4; while ((1 << shift) < dq) ++shift; }
    const int blocks = (int)((n4 + 255) / 256);
    lhr_apply_kernel<<<blocks, 256, 0, stream>>>(x, scale, out, n4, shift);
}